// BatchedModelManifoldGeodesicFlow_26757646254306
// MI455X (gfx1250) — compile-verified
//
#include <hip/hip_runtime.h>
#include <hip/hip_bf16.h>

// =====================================================================
// BatchedModelManifoldGeodesicFlow on gfx1250 (MI455X), f32 WMMA path.
//
// Math factorization (f(x) = tanh(xW1+b1)W2 + b2, J = W2^T diag(s) W1^T):
//   G      = W2^T D M D W2,  M = W1^T W1 (const),  D = diag(s), s=1-h^2
//   dG_ijk = T1[i,j,k] + T1[j,i,k],
//   T1[i,j,k] = sum_p X[p,i] Y[p,j] Z[p,k],
//     X = diag(c) W2 (c = -2 h s),  Y = R = M diag(s) W2,  Z = W1^T
//   ||dG||^2 = 2*sum_{pp'} (c c^T ∘ K2)∘(R R^T)∘M
//            + 2*sum_{pp'} Gxy[p,p'] Gxy[p',p] M[p,p'],
//     K2 = W2 W2^T (const), Gxy[p,p'] = c_p * Q[p',p], Q = R W2^T
//   Gamma contraction (with w=W2 v, alpha=c∘w, beta=M(s∘w), gamma=W1^T v):
//     S1_i = sum_p W2[p,i] c_p beta_p gamma_p + sum_p R[p,i] alpha_p gamma_p
//     S3_i = 2 sum_p W1[i,p] alpha_p beta_p
//     a_i  = -(S1_i - 0.5 S3_i) / (||dG||+1e-6) / (||v||+1e-6)
// =====================================================================

#define NN 128
#define HH 256
#define BB 16

typedef float v2f __attribute__((ext_vector_type(2)));
typedef float v8f __attribute__((ext_vector_type(8)));

// D = A(16x4) * B(4x16) + C, f32 WMMA (CDNA5 V_WMMA_F32_16X16X4_F32)
__device__ __forceinline__ v8f wmma4(v2f a, v2f b, v8f c) {
  return __builtin_amdgcn_wmma_f32_16x16x4_f32(
      /*neg_a=*/false, a, /*neg_b=*/false, b,
      /*c_mod=*/(short)0, c, /*reuse_a=*/false, /*reuse_b=*/false);
}

// ---------------------------------------------------------------------
// K1: per-sample prep. Block = 256 threads, grid = B.
// Computes s, c (H), w = W2 v, gamma = W1^T v, ||v||.
// ---------------------------------------------------------------------
__global__ void prep_kernel(const float* __restrict__ t_p,
                            const float* __restrict__ state,
                            const float* __restrict__ x0,
                            const float* __restrict__ x1,
                            const float* __restrict__ W1,
                            const float* __restrict__ b1,
                            const float* __restrict__ W2,
                            float* __restrict__ sbuf, float* __restrict__ cbuf,
                            float* __restrict__ wbuf, float* __restrict__ gbuf,
                            float* __restrict__ vnorm) {
  const int b = blockIdx.x, tid = threadIdx.x;
  __shared__ float lx[NN], lv[NN], red[HH];
  const float t = t_p[0];
  const float win = 4.0f * t * (1.0f - t);
  if (tid < NN) {
    const float dev = state[b * NN + tid];
    const float v = state[(BB + b) * NN + tid];
    const float a0 = x0[b * NN + tid];
    const float a1 = x1[b * NN + tid];
    lx[tid] = a0 + t * (a1 - a0) + win * dev;
    lv[tid] = v;
  }
  __syncthreads();
  // p = tid in [0,256)
  float z = b1[tid], g = 0.0f, w = 0.0f;
  for (int k = 0; k < NN; ++k) {
    const float w1 = W1[k * HH + tid];
    z += lx[k] * w1;
    g += lv[k] * w1;
  }
  for (int j = 0; j < NN; ++j) w += W2[tid * NN + j] * lv[j];
  const float h = tanhf(z);
  const float s = 1.0f - h * h;
  const float c = -2.0f * h * s;
  sbuf[b * HH + tid] = s;
  cbuf[b * HH + tid] = c;
  wbuf[b * HH + tid] = w;
  gbuf[b * HH + tid] = g;
  red[tid] = (tid < NN) ? lv[tid] * lv[tid] : 0.0f;
  __syncthreads();
  for (int off = HH / 2; off > 0; off >>= 1) {
    if (tid < off) red[tid] += red[tid + off];
    __syncthreads();
  }
  if (tid == 0) vnorm[b] = sqrtf(red[0]);
}

// ---------------------------------------------------------------------
// K2: constant Grams. grid = (16,16,2), block = 32 (one wave per tile).
//  z==0: M  = W1^T W1  (256x256, K=128)
//  z==1: K2 = W2 W2^T  (256x256, K=128)
// ---------------------------------------------------------------------
__global__ void const_gram_kernel(const float* __restrict__ W1,
                                  const float* __restrict__ W2,
                                  float* __restrict__ Mmat,
                                  float* __restrict__ K2mat) {
  const int l = threadIdx.x;
  const int tJ = blockIdx.x, tI = blockIdx.y, which = blockIdx.z;
  const int lane15 = l & 15;
  const int koff = (l < 16) ? 0 : 2;
  const int hi = (l < 16) ? 0 : 8;
  v8f acc = {};
  if (which == 0) {
    const int p = tI * 16 + lane15, q = tJ * 16 + lane15;
    for (int k0 = 0; k0 < NN; k0 += 4) {
      const int k = k0 + koff;
      v2f a, bv;
      a.x = W1[k * HH + p];       a.y = W1[(k + 1) * HH + p];
      bv.x = W1[k * HH + q];      bv.y = W1[(k + 1) * HH + q];
      acc = wmma4(a, bv, acc);
    }
    for (int r = 0; r < 8; ++r)
      Mmat[(tI * 16 + r + hi) * HH + tJ * 16 + lane15] = acc[r];
  } else {
    const int p = tI * 16 + lane15, q = tJ * 16 + lane15;
    for (int k0 = 0; k0 < NN; k0 += 4) {
      const int k = k0 + koff;
      v2f a, bv;
      a.x = W2[p * NN + k];       a.y = W2[p * NN + k + 1];
      bv.x = W2[q * NN + k];      bv.y = W2[q * NN + k + 1];
      acc = wmma4(a, bv, acc);
    }
    for (int r = 0; r < 8; ++r)
      K2mat[(tI * 16 + r + hi) * HH + tJ * 16 + lane15] = acc[r];
  }
}

// ---------------------------------------------------------------------
// K3: R[b] = M * (diag(s) W2), 256x128 out, K=256.
// grid = (8, 16, B), block = 32.
// ---------------------------------------------------------------------
__global__ void r_gemm_kernel(const float* __restrict__ Mmat,
                              const float* __restrict__ W2,
                              const float* __restrict__ sbuf,
                              float* __restrict__ Rbuf) {
  const int l = threadIdx.x;
  const int tJ = blockIdx.x, tI = blockIdx.y, b = blockIdx.z;
  __shared__ float sS[HH];
  for (int q = l; q < HH; q += 32) sS[q] = sbuf[b * HH + q];
  __syncthreads();
  const int lane15 = l & 15;
  const int koff = (l < 16) ? 0 : 2;
  const int hi = (l < 16) ? 0 : 8;
  const int p = tI * 16 + lane15;   // A row
  const int j = tJ * 16 + lane15;   // B col
  v8f acc = {};
  for (int k0 = 0; k0 < HH; k0 += 4) {
    const int q = k0 + koff;
    v2f a, bv;
    a.x = Mmat[p * HH + q];
    a.y = Mmat[p * HH + q + 1];
    bv.x = sS[q] * W2[q * NN + j];
    bv.y = sS[q + 1] * W2[(q + 1) * NN + j];
    acc = wmma4(a, bv, acc);
  }
  float* R = Rbuf + b * HH * NN;
  for (int r = 0; r < 8; ++r)
    R[(tI * 16 + r + hi) * NN + tJ * 16 + lane15] = acc[r];
}

// ---------------------------------------------------------------------
// K4: per-tile Gram contractions + norm^2 partial.
//  Gyy(I,J) = R R^T tile, Q(I,J) = R W2^T tile, QT = Q(J,I) tile.
//  partial  = sum M∘( c c' K2 ∘ Gyy + (c Q^T(J,I)) ∘ (c' Q(I,J)) )
// grid = (16, 16, B), block = 32.
// ---------------------------------------------------------------------
__global__ void gram_reduce_kernel(const float* __restrict__ Rbuf,
                                   const float* __restrict__ W2,
                                   const float* __restrict__ Mmat,
                                   const float* __restrict__ K2mat,
                                   const float* __restrict__ cbuf,
                                   float* __restrict__ partialBuf) {
  const int l = threadIdx.x;
  const int tJ = blockIdx.x, tI = blockIdx.y, b = blockIdx.z;
  const float* R = Rbuf + b * HH * NN;
  const int lane15 = l & 15;
  const int koff = (l < 16) ? 0 : 2;
  const int hi = (l < 16) ? 0 : 8;
  v8f accG = {}, accQ = {}, accQT = {};
  for (int k0 = 0; k0 < NN; k0 += 4) {
    const int k = k0 + koff;
    v2f aR, bR, bW, a2, bW2;
    aR.x = R[(tI * 16 + lane15) * NN + k];  aR.y = R[(tI * 16 + lane15) * NN + k + 1];
    bR.x = R[(tJ * 16 + lane15) * NN + k];  bR.y = R[(tJ * 16 + lane15) * NN + k + 1];
    bW.x = W2[(tJ * 16 + lane15) * NN + k]; bW.y = W2[(tJ * 16 + lane15) * NN + k + 1];
    a2.x = R[(tJ * 16 + lane15) * NN + k];  a2.y = R[(tJ * 16 + lane15) * NN + k + 1];
    bW2.x = W2[(tI * 16 + lane15) * NN + k]; bW2.y = W2[(tI * 16 + lane15) * NN + k + 1];
    accG = wmma4(aR, bR, accG);    // Gyy  = R(I) . R(J)^T
    accQ = wmma4(aR, bW, accQ);    // Q    = R(I) . W2(J)^T
    accQT = wmma4(a2, bW2, accQT); // Q^T tile: R(J) . W2(I)^T
  }
  // Stage QT tile (tile (J,I) layout) to LDS so we can read it transposed.
  __shared__ float qt[16][17];
  for (int r = 0; r < 8; ++r) qt[r + hi][lane15] = accQT[r];
  __syncthreads();
  const int pp = tJ * 16 + lane15;           // p' (column)
  const float cpp = cbuf[b * HH + pp];
  float partial = 0.0f;
  for (int r = 0; r < 8; ++r) {
    const int m = r + hi;
    const int p = tI * 16 + m;               // p (row)
    const float Mv = Mmat[p * HH + pp];
    const float K2v = K2mat[p * HH + pp];
    const float cp = cbuf[b * HH + p];
    // Q[p',p] lives in QT tile at local (row=p'-16J, col=p-16I) = qt[lane15][m]
    partial += Mv * cp * cpp * (K2v * accG[r] + accQ[r] * qt[lane15][m]);
  }
  for (int off = 16; off > 0; off >>= 1)
    partial += __shfl_xor(partial, off, 32);
  if (l == 0) partialBuf[b * HH + tI * 16 + tJ] = partial;
}

// ---------------------------------------------------------------------
// K5: final per-sample assembly. Block = 256, grid = B.
// ---------------------------------------------------------------------
__global__ void final_kernel(const float* __restrict__ state,
                             const float* __restrict__ W1,
                             const float* __restrict__ W2,
                             const float* __restrict__ Mmat,
                             const float* __restrict__ Rbuf,
                             const float* __restrict__ sbuf,
                             const float* __restrict__ cbuf,
                             const float* __restrict__ wbuf,
                             const float* __restrict__ gbuf,
                             const float* __restrict__ vnorm,
                             const float* __restrict__ partialBuf,
                             float* __restrict__ out) {
  const int b = blockIdx.x, tid = threadIdx.x;
  __shared__ float asw[HH], t1[HH], t2[HH], ab[HH], red[HH];
  const float s = sbuf[b * HH + tid];
  const float c = cbuf[b * HH + tid];
  const float w = wbuf[b * HH + tid];
  const float g = gbuf[b * HH + tid];
  const float alpha = c * w;
  asw[tid] = s * w;
  red[tid] = partialBuf[b * HH + tid];
  __syncthreads();
  float beta = 0.0f;
  for (int q = 0; q < HH; ++q) beta += Mmat[tid * HH + q] * asw[q];
  t1[tid] = c * beta * g;       // for sum_p W2[p,i] * (c beta gamma)_p
  t2[tid] = alpha * g;          // for sum_p R[p,i]  * (alpha gamma)_p
  ab[tid] = alpha * beta;       // for S3
  for (int off = HH / 2; off > 0; off >>= 1) {
    if (tid < off) red[tid] += red[tid + off];
    __syncthreads();
  }
  const float nrm = sqrtf(2.0f * red[0]);
  const float vn = vnorm[b];
  if (tid < NN) {
    const int i = tid;
    const float* R = Rbuf + b * HH * NN;
    float S1 = 0.0f, S3 = 0.0f;
    for (int p = 0; p < HH; ++p) {
      S1 += W2[p * NN + i] * t1[p] + R[p * NN + i] * t2[p];
      S3 += W1[i * HH + p] * ab[p];
    }
    S3 *= 2.0f;
    const float contr = S1 - 0.5f * S3;
    float ai = -contr / (nrm + 1e-6f);
    ai = ai / (vn + 1e-6f);
    const float dev = state[b * NN + i];
    const float v = state[(BB + b) * NN + i];
    out[b * NN + i] = v;
    out[(BB + b) * NN + i] = ai - 0.1f * dev;
  }
}

// ---------------------------------------------------------------------
extern "C" void kernel_launch(void* const* d_in, const int* in_sizes, int n_in,
                              void* d_out, int out_size, void* d_ws, size_t ws_size,
                              hipStream_t stream) {
  (void)in_sizes; (void)n_in; (void)out_size; (void)ws_size;
  const float* t_p   = (const float*)d_in[0];
  const float* state = (const float*)d_in[1];
  const float* x0    = (const float*)d_in[2];
  const float* x1    = (const float*)d_in[3];
  const float* W1    = (const float*)d_in[4];
  const float* b1    = (const float*)d_in[5];
  const float* W2    = (const float*)d_in[6];
  // d_in[7] = b2: unused (does not affect J or dG)
  float* out = (float*)d_out;

  // Workspace layout (floats)
  float* ws = (float*)d_ws;
  float* Mmat    = ws;                  // 256*256
  float* K2mat   = Mmat + HH * HH;      // 256*256
  float* sbuf    = K2mat + HH * HH;     // B*256
  float* cbuf    = sbuf + BB * HH;      // B*256
  float* wbuf    = cbuf + BB * HH;      // B*256
  float* gbuf    = wbuf + BB * HH;      // B*256
  float* vnorm   = gbuf + BB * HH;      // B
  float* partial = vnorm + BB;          // B*256
  float* Rbuf    = partial + BB * HH;   // B*256*128

  prep_kernel<<<BB, HH, 0, stream>>>(t_p, state, x0, x1, W1, b1, W2,
                                     sbuf, cbuf, wbuf, gbuf, vnorm);
  const_gram_kernel<<<dim3(16, 16, 2), 32, 0, stream>>>(W1, W2, Mmat, K2mat);
  r_gemm_kernel<<<dim3(8, 16, BB), 32, 0, stream>>>(Mmat, W2, sbuf, Rbuf);
  gram_reduce_kernel<<<dim3(16, 16, BB), 32, 0, stream>>>(Rbuf, W2, Mmat, K2mat,
                                                          cbuf, partial);
  final_kernel<<<BB, HH, 0, stream>>>(state, W1, W2, Mmat, Rbuf, sbuf, cbuf,
                                      wbuf, gbuf, vnorm, partial, out);
}